// JustLightGCN_80393197846613
// MI455X (gfx1250) — compile-verified
//
#include <hip/hip_runtime.h>
#include <hip/hip_bf16.h>
#include <stdint.h>

#define DIM 128          // feature dim (reference D)
#define TILE 256         // edges staged per block
#define BLK 256          // threads per block (8 wave32s)
#define WAVES (BLK / 32)
#define PIPE 4           // async row-gather pipeline depth (LDS buffers per wave)

#if __has_builtin(__builtin_amdgcn_global_load_async_to_lds_b32) && \
    __has_builtin(__builtin_amdgcn_global_load_async_to_lds_b128)
#define USE_ASYNC 1
#else
#define USE_ASYNC 0
#endif

typedef int v4i __attribute__((ext_vector_type(4)));
typedef __attribute__((address_space(1))) int* g_i32p;   // global (AS1)
typedef __attribute__((address_space(3))) int* l_i32p;   // LDS (AS3)
typedef __attribute__((address_space(1))) v4i* g_v4ip;
typedef __attribute__((address_space(3))) v4i* l_v4ip;

__device__ __forceinline__ void async_wait_all() {
#if __has_builtin(__builtin_amdgcn_s_wait_asynccnt)
    __builtin_amdgcn_s_wait_asynccnt(0);
#else
    asm volatile("s_wait_asynccnt 0" ::: "memory");
#endif
}
__device__ __forceinline__ void async_wait_pipe() {
#if __has_builtin(__builtin_amdgcn_s_wait_asynccnt)
    __builtin_amdgcn_s_wait_asynccnt(PIPE - 1);
#else
    asm volatile("s_wait_asynccnt 3" ::: "memory");
#endif
}

// single-instruction f32 atomic add, no return (global_atomic_add_f32)
__device__ __forceinline__ void atomic_add_f32(float* p, float v) {
    __hip_atomic_fetch_add(p, v, __ATOMIC_RELAXED, __HIP_MEMORY_SCOPE_AGENT);
}

// ---------------- propagation: y[col] += x[row] * norm[edge] ----------------
// One wave32 per edge (32 lanes x float4 = one 128-float feature row).
// row/col/norm tiles staged into LDS via async global->LDS; each wave keeps
// PIPE-1 random row gathers in flight via global_load_async_to_lds_b128
// (ASYNCcnt-tracked). Steady-state loop is branch-free with a constant
// s_wait_asynccnt PIPE-1; the epilogue drains once and finishes without waits.

__global__ __launch_bounds__(BLK)
void k_prop(const float* __restrict__ x, const int* __restrict__ row,
            const int* __restrict__ col, const float* __restrict__ nrm_e,
            float* __restrict__ y, int e)
{
    __shared__ int   s_row[TILE];
    __shared__ int   s_col[TILE];
    __shared__ float s_nrm[TILE];
#if USE_ASYNC
    __shared__ __align__(16) float s_x[WAVES][PIPE][DIM];
#endif

    const int base = blockIdx.x * TILE;
    const int tid  = threadIdx.x;
    const int nvalid = (e - base < TILE) ? (e - base) : TILE;

#if USE_ASYNC
    if (tid < nvalid) {
        __builtin_amdgcn_global_load_async_to_lds_b32(
            (g_i32p)(row + base + tid), (l_i32p)&s_row[tid], 0, 0);
        __builtin_amdgcn_global_load_async_to_lds_b32(
            (g_i32p)(col + base + tid), (l_i32p)&s_col[tid], 0, 0);
        __builtin_amdgcn_global_load_async_to_lds_b32(
            (g_i32p)(nrm_e + base + tid), (l_i32p)&s_nrm[tid], 0, 0);
    }
    async_wait_all();
#else
    if (tid < nvalid) {
        s_row[tid] = row[base + tid];
        s_col[tid] = col[base + tid];
        s_nrm[tid] = nrm_e[base + tid];
    }
#endif
    __syncthreads();

    const int wave  = tid >> 5;          // 8 waves/block
    const int lane  = tid & 31;          // wave32
    const int wbase = base + wave * 32;  // this wave's first edge
    int cnt = e - wbase;
    if (cnt < 0)  cnt = 0;
    if (cnt > 32) cnt = 32;

#if USE_ASYNC
    // prologue: fill the pipeline with up to PIPE-1 gathers
    const int pre = (cnt < PIPE - 1) ? cnt : (PIPE - 1);
    for (int j = 0; j < pre; ++j) {
        const int rj = s_row[wave * 32 + j];
        __builtin_amdgcn_global_load_async_to_lds_b128(
            (g_v4ip)(x + (size_t)rj * DIM + (size_t)lane * 4),
            (l_v4ip)&s_x[wave][j][lane * 4], 0, 0);
    }

    // steady state: issue k+PIPE-1, wait to <=PIPE-1 outstanding, consume k
    const int main_cnt = cnt - (PIPE - 1);
    int k = 0;
    for (; k < main_cnt; ++k) {
        const int el = wave * 32 + k;

        const int rn = s_row[el + PIPE - 1];
        __builtin_amdgcn_global_load_async_to_lds_b128(
            (g_v4ip)(x + (size_t)rn * DIM + (size_t)lane * 4),
            (l_v4ip)&s_x[wave][(k + PIPE - 1) & (PIPE - 1)][lane * 4], 0, 0);
        async_wait_pipe();   // gather k has landed (in-order completion)

        const float  nrm = s_nrm[el];
        const int    c   = s_col[el];
        const float4 v   = *(const float4*)&s_x[wave][k & (PIPE - 1)][lane * 4];
        float* dst = y + (size_t)c * DIM + (size_t)lane * 4;
        atomic_add_f32(dst + 0, v.x * nrm);
        atomic_add_f32(dst + 1, v.y * nrm);
        atomic_add_f32(dst + 2, v.z * nrm);
        atomic_add_f32(dst + 3, v.w * nrm);
    }

    // epilogue: all remaining gathers already issued; drain once, no more waits
    if (k < cnt) {
        async_wait_all();
        for (; k < cnt; ++k) {
            const int el = wave * 32 + k;
            const float  nrm = s_nrm[el];
            const int    c   = s_col[el];
            const float4 v   = *(const float4*)&s_x[wave][k & (PIPE - 1)][lane * 4];
            float* dst = y + (size_t)c * DIM + (size_t)lane * 4;
            atomic_add_f32(dst + 0, v.x * nrm);
            atomic_add_f32(dst + 1, v.y * nrm);
            atomic_add_f32(dst + 2, v.z * nrm);
            atomic_add_f32(dst + 3, v.w * nrm);
        }
    }
#else
    for (int k = 0; k < cnt; ++k) {
        const int el = wave * 32 + k;
        const int r  = s_row[el];
        const int c  = s_col[el];
        const float nrm = s_nrm[el];
        if (k + 1 < cnt) {
            __builtin_prefetch(x + (size_t)s_row[el + 1] * DIM + (size_t)lane * 4, 0, 3);
        }
        const float4 v = *(const float4*)(x + (size_t)r * DIM + (size_t)lane * 4);
        float* dst = y + (size_t)c * DIM + (size_t)lane * 4;
        atomic_add_f32(dst + 0, v.x * nrm);
        atomic_add_f32(dst + 1, v.y * nrm);
        atomic_add_f32(dst + 2, v.z * nrm);
        atomic_add_f32(dst + 3, v.w * nrm);
    }
#endif
}

// ---------------- utility kernels ----------------

__global__ void k_zero(float* __restrict__ p, int n) {
    int i = blockIdx.x * blockDim.x + threadIdx.x;
    if (i < n) p[i] = 0.0f;
}

// integer degree count (exact, always lowers to HW atomic)
__global__ void k_deg(const int* __restrict__ col, unsigned int* __restrict__ deg, int e) {
    int i = blockIdx.x * blockDim.x + threadIdx.x;
    if (i < e) atomicAdd(&deg[col[i]], 1u);
}

// deg (as uint bits) -> 1/sqrt(deg) in place
__global__ void k_rsqrt(float* __restrict__ d, int n) {
    int i = blockIdx.x * blockDim.x + threadIdx.x;
    if (i < n) {
        unsigned int v = __float_as_uint(d[i]);
        d[i] = (v > 0u) ? rsqrtf((float)v) : 0.0f;
    }
}

// per-edge symmetric norm, computed once and reused by both layers
__global__ void k_norm(const int* __restrict__ row, const int* __restrict__ col,
                       const float* __restrict__ dinv, float* __restrict__ nrm, int e) {
    int i = blockIdx.x * blockDim.x + threadIdx.x;
    if (i < e) nrm[i] = dinv[row[i]] * dinv[col[i]];
}

// out = a + b   (float4 vectorized; N*D divisible by 4)
__global__ void k_combine4(const float4* __restrict__ a, const float4* __restrict__ b,
                           float4* __restrict__ o, int n4) {
    int i = blockIdx.x * blockDim.x + threadIdx.x;
    if (i < n4) {
        float4 xx = a[i], yy = b[i];
        float4 r; r.x = xx.x + yy.x; r.y = xx.y + yy.y; r.z = xx.z + yy.z; r.w = xx.w + yy.w;
        o[i] = r;
    }
}

__global__ void k_scale4(float4* __restrict__ o, float s, int n4) {
    int i = blockIdx.x * blockDim.x + threadIdx.x;
    if (i < n4) {
        float4 v = o[i];
        v.x *= s; v.y *= s; v.z *= s; v.w *= s;
        o[i] = v;
    }
}

// ---------------- host-side launch ----------------

extern "C" void kernel_launch(void* const* d_in, const int* in_sizes, int n_in,
                              void* d_out, int out_size, void* d_ws, size_t ws_size,
                              hipStream_t stream)
{
    const float* emb = (const float*)d_in[0];
    const int*   ei  = (const int*)  d_in[1];

    const int nd = in_sizes[0];        // N * D
    const int n  = nd / DIM;           // N
    const int e  = in_sizes[1] / 2;    // E
    const int* row = ei;               // edge_index[0, :]
    const int* col = ei + e;           // edge_index[1, :]

    float* out  = (float*)d_out;
    float* dinv = (float*)d_ws;                       // N floats (deg counts, then rsqrt)
    float* nrm  = dinv + ((n + 255) & ~255);          // E floats (per-edge norm)
    float* x1   = nrm  + ((e + 255) & ~255);          // N*D floats

    const int n4 = nd / 4;

    // degree -> dinv -> per-edge norm
    k_zero <<<(n  + BLK - 1) / BLK, BLK, 0, stream>>>(dinv, n);
    k_zero <<<(nd + BLK - 1) / BLK, BLK, 0, stream>>>(x1, nd);
    k_deg  <<<(e  + BLK - 1) / BLK, BLK, 0, stream>>>(col, (unsigned int*)dinv, e);
    k_rsqrt<<<(n  + BLK - 1) / BLK, BLK, 0, stream>>>(dinv, n);
    k_norm <<<(e  + BLK - 1) / BLK, BLK, 0, stream>>>(row, col, dinv, nrm, e);

    const int prop_blocks = (e + TILE - 1) / TILE;

    // layer 1: x1 = scatter(emb)
    k_prop<<<prop_blocks, BLK, 0, stream>>>(emb, row, col, nrm, x1, e);

    // out = emb + x1   (alpha0 + alpha1 terms, unnormalized)
    k_combine4<<<(n4 + BLK - 1) / BLK, BLK, 0, stream>>>(
        (const float4*)emb, (const float4*)x1, (float4*)out, n4);

    // layer 2: scatter(x1) accumulated directly into out (saves a buffer + pass)
    k_prop<<<prop_blocks, BLK, 0, stream>>>(x1, row, col, nrm, out, e);

    // out /= (NUM_LAYERS + 1)
    k_scale4<<<(n4 + BLK - 1) / BLK, BLK, 0, stream>>>((float4*)out, 1.0f / 3.0f, n4);
}